// QuantumAttentionNet_1082331759394
// MI455X (gfx1250) — compile-verified
//
#include <hip/hip_runtime.h>
#include <math.h>

#define NROWS 8192
#define DIM   144
#define DIMP  148     // padded LDS row stride (dwords): 16-lane stride-148 reads are bank-conflict-free
#define OUTC  15

typedef float v2f __attribute__((ext_vector_type(2)));
typedef float v8f __attribute__((ext_vector_type(8)));
typedef int   v4i __attribute__((ext_vector_type(4)));

#if __has_builtin(__builtin_amdgcn_global_load_async_to_lds_b128) && \
    __has_builtin(__builtin_amdgcn_s_wait_asynccnt)
#define HAVE_ASYNC_LDS 1
#else
#define HAVE_ASYNC_LDS 0
#endif

// address-space-qualified pointee typedefs for the async builtin
typedef __attribute__((address_space(1))) v4i g_v4i;   // global
typedef __attribute__((address_space(3))) v4i l_v4i;   // LDS

__device__ __forceinline__ void stage16(void* lds, const void* g) {
#if HAVE_ASYNC_LDS
  // async DMA: global -> LDS, no VGPR round trip, tracked by ASYNCcnt
  __builtin_amdgcn_global_load_async_to_lds_b128((g_v4i*)g, (l_v4i*)lds, 0, 0);
#else
  *(float4*)lds = *(const float4*)g;
#endif
}

__device__ __forceinline__ void stage_wait() {
#if HAVE_ASYNC_LDS
  __builtin_amdgcn_s_wait_asynccnt(0);
#endif
}

__device__ __forceinline__ v8f wmma_f32_16x16x4(v2f a, v2f b, v8f c) {
  return __builtin_amdgcn_wmma_f32_16x16x4_f32(false, a, false, b, (short)0, c,
                                               false, false);
}

// ---------------------------------------------------------------------------
// Kernel 1: Q/K/V projections.  Y = x @ W^T + b  (per blockIdx.z matrix)
// One wave computes one 16x16 output tile with 36 f32 WMMA steps (K=144).
// ---------------------------------------------------------------------------
__global__ __launch_bounds__(32) void qkv_kernel(
    const float* __restrict__ x,
    const float* __restrict__ Wq, const float* __restrict__ bq,
    const float* __restrict__ Wk, const float* __restrict__ bk,
    const float* __restrict__ Wv, const float* __restrict__ bv,
    float* __restrict__ Q, float* __restrict__ K, float* __restrict__ V) {
  const int lane  = threadIdx.x;
  const int l15   = lane & 15;
  const int half  = lane >> 4;
  const int mbase = blockIdx.x * 16;
  const int nbase = blockIdx.y * 16;
  const int mat   = blockIdx.z;

  const float* W  = (mat == 0) ? Wq : (mat == 1) ? Wk : Wv;
  const float* bs = (mat == 0) ? bq : (mat == 1) ? bk : bv;
  float*       Y  = (mat == 0) ? Q  : (mat == 1) ? K  : V;

  v8f acc = {};
#pragma unroll
  for (int kk = 0; kk < DIM; kk += 4) {
    const int kc = kk + half * 2;
    v2f a = *(const v2f*)(x + (size_t)(mbase + l15) * DIM + kc);
    v2f b = *(const v2f*)(W + (size_t)(nbase + l15) * DIM + kc);
    acc = wmma_f32_16x16x4(a, b, acc);
  }
  const float bias = bs[nbase + l15];
#pragma unroll
  for (int r = 0; r < 8; ++r) {
    const int row = mbase + half * 8 + r;
    Y[(size_t)row * DIM + nbase + l15] = acc[r] + bias;
  }
}

// ---------------------------------------------------------------------------
// Kernel 2: flash attention.  4 waves/block, each owns a 16-row query tile.
// K/V tiles are staged once per block into LDS via async-to-LDS DMA,
// cutting global traffic 4x and making the stage fully coalesced.
// ---------------------------------------------------------------------------
__global__ __launch_bounds__(128) void attn_kernel(
    const float* __restrict__ Q, const float* __restrict__ K,
    const float* __restrict__ V, float* __restrict__ Aout) {
  __shared__ __align__(16) float Klds[16][DIMP];
  __shared__ __align__(16) float Vlds[16][DIMP];
  __shared__ float plds[4][16][17];

  const int tid   = threadIdx.x;
  const int lane  = tid & 31;
  const int wave  = tid >> 5;
  const int l15   = lane & 15;
  const int half  = lane >> 4;
  const int mbase = (blockIdx.x * 4 + wave) * 16;

  // Preload this wave's Q tile in A-layout: 36 K-chunks of 2 floats per lane.
  v2f aq[36];
#pragma unroll
  for (int c = 0; c < 36; ++c)
    aq[c] = *(const v2f*)(Q + (size_t)(mbase + l15) * DIM + c * 4 + half * 2);

  v8f   o[9];
  float m[8], l[8];
#pragma unroll
  for (int j = 0; j < 9; ++j) o[j] = (v8f){};
#pragma unroll
  for (int r = 0; r < 8; ++r) { m[r] = -INFINITY; l[r] = 0.0f; }

  const float scale = 0.083333333333333333f;  // 1/sqrt(144)

  for (int kb = 0; kb < NROWS; kb += 16) {
    // ---- cooperative stage of K/V tiles: 576 x 16B chunks each ----
    {
      const float4* gk = (const float4*)(K + (size_t)kb * DIM);
      const float4* gv = (const float4*)(V + (size_t)kb * DIM);
#pragma unroll
      for (int i = 0; i < 5; ++i) {
        const int idx = tid + i * 128;          // 0..575 (tile is 16*36 chunks)
        if (idx < 576) {
          const int r  = idx / 36;              // tile row
          const int c4 = idx % 36;              // float4 within row
          stage16(&Klds[r][c4 * 4], gk + idx);
          stage16(&Vlds[r][c4 * 4], gv + idx);
        }
      }
      stage_wait();
      __syncthreads();
    }

    // ---- S tile = Q_tile @ K_tile^T  (B operand from LDS) ----
    v8f s = {};
#pragma unroll
    for (int c = 0; c < 36; ++c) {
      v2f b = *(const v2f*)&Klds[l15][c * 4 + half * 2];
      s = wmma_f32_16x16x4(aq[c], b, s);
    }

    // ---- online softmax (rows split per 16-lane half, matching C layout) ----
    float fac[8];
#pragma unroll
    for (int r = 0; r < 8; ++r) {
      float sv = s[r] * scale;
      float mx = sv;
      mx = fmaxf(mx, __shfl_xor(mx, 1, 32));
      mx = fmaxf(mx, __shfl_xor(mx, 2, 32));
      mx = fmaxf(mx, __shfl_xor(mx, 4, 32));
      mx = fmaxf(mx, __shfl_xor(mx, 8, 32));
      const float mn = fmaxf(m[r], mx);
      fac[r] = __expf(m[r] - mn);
      m[r]   = mn;
      const float p = __expf(sv - mn);
      float ts = p;
      ts += __shfl_xor(ts, 1, 32);
      ts += __shfl_xor(ts, 2, 32);
      ts += __shfl_xor(ts, 4, 32);
      ts += __shfl_xor(ts, 8, 32);
      l[r] = l[r] * fac[r] + ts;
      s[r] = p;                      // P tile now lives in C layout
    }
#pragma unroll
    for (int j = 0; j < 9; ++j)
#pragma unroll
      for (int r = 0; r < 8; ++r) o[j][r] *= fac[r];

    // ---- relayout P (C layout -> A layout) via per-wave LDS buffer ----
#pragma unroll
    for (int r = 0; r < 8; ++r) plds[wave][half * 8 + r][l15] = s[r];
    asm volatile("s_wait_dscnt 0" ::: "memory");
    v2f pa[4];
#pragma unroll
    for (int c = 0; c < 4; ++c) {
      pa[c].x = plds[wave][l15][c * 4 + half * 2 + 0];
      pa[c].y = plds[wave][l15][c * 4 + half * 2 + 1];
    }

    // ---- O += P @ V_tile  (V from LDS; K=16 -> 4 WMMA steps per slab) ----
#pragma unroll
    for (int j = 0; j < 9; ++j) {
      v8f acc = o[j];
#pragma unroll
      for (int c = 0; c < 4; ++c) {
        const int kr = c * 4 + half * 2;
        v2f b;
        b.x = Vlds[kr + 0][j * 16 + l15];
        b.y = Vlds[kr + 1][j * 16 + l15];
        acc = wmma_f32_16x16x4(pa[c], b, acc);
      }
      o[j] = acc;
    }

    __syncthreads();   // protect K/V tiles from next iteration's staging
  }

  // ---- normalize and store ----
#pragma unroll
  for (int r = 0; r < 8; ++r) {
    const float inv = 1.0f / l[r];
    const int   row = mbase + half * 8 + r;
#pragma unroll
    for (int j = 0; j < 9; ++j)
      Aout[(size_t)row * DIM + j * 16 + l15] = o[j][r] * inv;
  }
}

// ---------------------------------------------------------------------------
// Kernel 3: head.  Only h[:, :4] feeds the quantum block.
// ---------------------------------------------------------------------------
__global__ __launch_bounds__(256) void head_kernel(
    const float* __restrict__ Aout, const float* __restrict__ W1,
    const float* __restrict__ b1, const float* __restrict__ W2,
    const float* __restrict__ b2, float* __restrict__ out) {
  const int row = blockIdx.x * 256 + threadIdx.x;
  if (row >= NROWS) return;
  const float* a = Aout + (size_t)row * DIM;

  float p0[4], p1[4];
#pragma unroll
  for (int j = 0; j < 4; ++j) {
    float s = b1[j];
    for (int k = 0; k < DIM; ++k) s = fmaf(a[k], W1[j * DIM + k], s);
    const float th = fmaxf(s, 0.0f);
    const float c  = cosf(0.5f * th);
    p0[j] = c * c;
    p1[j] = 1.0f - p0[j];
  }

  const int invperm[16] = {0, 1, 3, 2, 6, 7, 5, 4, 12, 13, 15, 14, 10, 11, 9, 8};
  float q[4] = {0.f, 0.f, 0.f, 0.f};
#pragma unroll
  for (int j = 0; j < 16; ++j) {
    const int   idx = invperm[j];
    const float st  = ((idx & 8) ? p1[0] : p0[0]) * ((idx & 4) ? p1[1] : p0[1]) *
                      ((idx & 2) ? p1[2] : p0[2]) * ((idx & 1) ? p1[3] : p0[3]);
    q[0] += (j & 8) ? -st : st;
    q[1] += (j & 4) ? -st : st;
    q[2] += (j & 2) ? -st : st;
    q[3] += (j & 1) ? -st : st;
  }

#pragma unroll
  for (int oc = 0; oc < OUTC; ++oc) {
    float s = b2[oc];
#pragma unroll
    for (int k = 0; k < 4; ++k) s = fmaf(q[k], W2[oc * 4 + k], s);
    out[(size_t)row * OUTC + oc] = s;
  }
}

// ---------------------------------------------------------------------------
extern "C" void kernel_launch(void* const* d_in, const int* in_sizes, int n_in,
                              void* d_out, int out_size, void* d_ws,
                              size_t ws_size, hipStream_t stream) {
  const float* x  = (const float*)d_in[0];
  const float* Wq = (const float*)d_in[1];
  const float* bq = (const float*)d_in[2];
  const float* Wk = (const float*)d_in[3];
  const float* bk = (const float*)d_in[4];
  const float* Wv = (const float*)d_in[5];
  const float* bv = (const float*)d_in[6];
  const float* W1 = (const float*)d_in[7];
  const float* b1 = (const float*)d_in[8];
  // d_in[9] = qw (unused by the reference math)
  const float* W2 = (const float*)d_in[10];
  const float* b2 = (const float*)d_in[11];

  const size_t mat_elems = (size_t)NROWS * DIM;
  float* Q = (float*)d_ws;
  float* K = Q + mat_elems;
  float* V = K + mat_elems;
  float* A = V + mat_elems;

  dim3 g1(NROWS / 16, DIM / 16, 3);
  qkv_kernel<<<g1, 32, 0, stream>>>(x, Wq, bq, Wk, bk, Wv, bv, Q, K, V);

  attn_kernel<<<NROWS / 64, 128, 0, stream>>>(Q, K, V, A);

  head_kernel<<<NROWS / 256, 256, 0, stream>>>(A, W1, b1, W2, b2,
                                               (float*)d_out);
}